// SPP_343597384129
// MI455X (gfx1250) — compile-verified
//
#include <hip/hip_runtime.h>

// ---------------------------------------------------------------------------
// SPP (k=5/9/13, stride 1, "same") on x[16,512,32,32] f32 -> out[16,2048,32,32]
// SPPF cascade: pool9 = pool5(pool5(x)), pool13 = pool5(pool9). Separable 1-D
// 5-max passes through LDS with -inf halos. Async global->LDS loads (ASYNCcnt)
// double-buffer the next plane while the current one is pooled.
// ---------------------------------------------------------------------------

#if __has_builtin(__builtin_amdgcn_global_load_async_to_lds_b128) && \
    __has_builtin(__builtin_amdgcn_s_wait_asynccnt)
#define HAVE_ASYNC 1
#else
#define HAVE_ASYNC 0
#endif

typedef float v4f __attribute__((ext_vector_type(4)));
typedef int   v4i __attribute__((__vector_size__(4 * sizeof(int))));  // matches builtin proto
typedef __attribute__((address_space(3))) v4i lv4i;                   // v4i in LDS (AS3)

static constexpr int kC          = 512;
static constexpr int kPlanes     = 16 * 512;     // 8192 (B*C)
static constexpr int kPlaneElems = 32 * 32;      // 1024
static constexpr int PPB         = 4;            // planes per block
static constexpr int STRA        = 36;           // 32 data + 4 trailing -inf per row
static constexpr int SZA         = 4 + 32 * STRA;// + 4 leading -inf floats
static constexpr int STRB        = 33;           // odd stride: conflict-free column reads
static constexpr int SZB         = 36 * STRB;    // rows -2..33 (2 -inf halo rows each end)

__global__ __launch_bounds__(256) void spp_sppf_kernel(const float* __restrict__ x,
                                                       float* __restrict__ out) {
  __shared__ __align__(16) float Abuf[2][SZA];   // ping-pong plane buffers
  __shared__ __align__(16) float Bbuf[SZB];      // horizontal-pass result

  const int tid = threadIdx.x;
  const int r   = tid >> 3;          // row 0..31
  const int c0  = (tid & 7) << 2;    // col chunk 0,4,...,28
  const int lA  = 4 + r * STRA + c0; // this thread's float4 slot in an A buffer (16B aligned)

  // ---- one-time -inf halo init (halo cells are never overwritten) ----
  for (int q = tid; q < 2 * 132 + 4 * STRB; q += 256) {
    if (q < 264) {                               // A buffers: 4 leading + 4 per row
      const int buf = (q >= 132) ? 1 : 0;
      const int e   = q - buf * 132;
      const int idx = (e < 4) ? e : (4 + ((e - 4) >> 2) * STRA + 32 + ((e - 4) & 3));
      Abuf[buf][idx] = -__builtin_inff();
    } else {                                     // B buffer: halo rows 0,1,34,35
      const int e  = q - 264;
      int row = e / STRB;
      row += (row >= 2) ? 32 : 0;
      Bbuf[row * STRB + e % STRB] = -__builtin_inff();
    }
  }

  const int p0 = blockIdx.x * PPB;

#if HAVE_ASYNC
  // prologue: kick off async copy of first plane into slot 0
  __builtin_amdgcn_global_load_async_to_lds_b128(
      (v4i*)(x + (size_t)p0 * kPlaneElems + tid * 4),
      (lv4i*)&Abuf[0][lA], 0, 0);
#endif

  for (int i = 0; i < PPB; ++i) {
    const int p = p0 + i;
    const int b = p >> 9;            // p / 512
    const int c = p & 511;           // p % 512
    float* A = Abuf[i & 1];
    float* dstBase = out + ((size_t)b * (4 * kC) + c) * kPlaneElems;

#if HAVE_ASYNC
    __builtin_amdgcn_s_wait_asynccnt(0);   // this wave's portion landed in LDS
    __syncthreads();                       // all waves' portions landed
    if (i + 1 < PPB) {                     // prefetch next plane into other slot
      __builtin_amdgcn_global_load_async_to_lds_b128(
          (v4i*)(x + (size_t)(p + 1) * kPlaneElems + tid * 4),
          (lv4i*)&Abuf[(i + 1) & 1][lA], 0, 0);
    }
#else
    const v4f vin = ((const v4f*)(x + (size_t)p * kPlaneElems))[tid];
    *(v4f*)&A[lA] = vin;
    __syncthreads();
#endif

    // identity channel group (copy of x)
    {
      const v4f v = *(const v4f*)&A[lA];
      __builtin_nontemporal_store(v, (v4f*)(dstBase + tid * 4));
    }

    // three cascaded 5x5 pools: A holds current field, result written back to A
    #pragma unroll
    for (int s = 0; s < 3; ++s) {
      // ---- horizontal 5-max: A -> B (halo makes edges branch-free) ----
      float w[8];
      #pragma unroll
      for (int j = 0; j < 8; ++j) w[j] = A[lA - 2 + j];
      #pragma unroll
      for (int j = 0; j < 4; ++j) {
        const float m = fmaxf(fmaxf(fmaxf(w[j], w[j + 1]),
                                    fmaxf(w[j + 2], w[j + 3])), w[j + 4]);
        Bbuf[(r + 2) * STRB + c0 + j] = m;
      }
      __syncthreads();

      // ---- vertical 5-max: B -> registers ----
      v4f o;
      #pragma unroll
      for (int j = 0; j < 4; ++j) {
        float m = Bbuf[r * STRB + c0 + j];
        #pragma unroll
        for (int d = 1; d < 5; ++d) m = fmaxf(m, Bbuf[(r + d) * STRB + c0 + j]);
        o[j] = m;
      }
      *(v4f*)&A[lA] = o;   // feed next cascade stage (5∘5=9, 5∘5∘5=13)

      float* dstj = dstBase + (size_t)(s + 1) * kC * kPlaneElems;
      __builtin_nontemporal_store(o, (v4f*)(dstj + tid * 4));
      __syncthreads();     // B reads + A writeback done before next H pass / plane
    }
  }
}

extern "C" void kernel_launch(void* const* d_in, const int* in_sizes, int n_in,
                              void* d_out, int out_size, void* d_ws, size_t ws_size,
                              hipStream_t stream) {
  (void)in_sizes; (void)n_in; (void)out_size; (void)d_ws; (void)ws_size;
  const float* x  = (const float*)d_in[0];
  float* out      = (float*)d_out;
  spp_sppf_kernel<<<dim3(kPlanes / PPB), dim3(256), 0, stream>>>(x, out);
}